// CannyNet_84739704750345
// MI455X (gfx1250) — compile-verified
//
#include <hip/hip_runtime.h>
#include <math.h>

typedef __attribute__((ext_vector_type(2))) float v2f;
typedef __attribute__((ext_vector_type(8))) float v8f;

#define IH 1024
#define IW 1024

// Unnormalized gaussian taps exp(-0.5*(i-2)^2), i=0..4 (matches reference)
__device__ __forceinline__ float bandv(int m) {
    // returns g[m] for m in [0,4], else 0  (g = {e^-2, e^-.5, 1, e^-.5, e^-2})
    float r = 0.0f;
    r = (m == 0 || m == 4) ? 0.13533528f : r;
    r = (m == 1 || m == 3) ? 0.60653066f : r;
    r = (m == 2)           ? 1.0f        : r;
    return r;
}

// ---------------------------------------------------------------------------
// Kernel 1: horizontal 5-tap blur via WMMA.
// One wave per 16x16 tile: Out(16x16) = In(16x20) * S(20x16), S banded.
// A (16x4 f32, 2 VGPRs): lane L<16 holds M=L,K={0,1}; lane L+16 holds M=L,K={2,3}.
// B (4x16 f32, 2 VGPRs): lane n<16 holds N=n,K={0,1}? -> row-striped: v0={K0,K2}, v1={K1,K3}.
// ---------------------------------------------------------------------------
__global__ __launch_bounds__(256) void k_hblur(const float* __restrict__ img,
                                               float* __restrict__ hb) {
    const int lane = threadIdx.x & 31;
    const int wave = threadIdx.x >> 5;
    const int tile = blockIdx.x * 8 + wave;
    const int b    = tile >> 12;          // 4096 tiles per 1024x1024 image
    const int rem  = tile & 4095;
    const int y0   = (rem >> 6) << 4;
    const int x0   = (rem & 63) << 4;

    const float* src = img + (size_t)b * (IH * IW);
    const int row = y0 + (lane & 15);     // M index
    const int cb  = (lane >> 4) << 1;     // K sub-offset: 0 or 2
    const int j   = lane & 15;            // N index for B operand

    v8f acc = {0.f, 0.f, 0.f, 0.f, 0.f, 0.f, 0.f, 0.f};
#pragma unroll
    for (int c = 0; c < 5; ++c) {
        const int k0  = 4 * c + cb;
        const int col = x0 - 2 + k0;
        float a0 = 0.f, a1 = 0.f;
        if ((unsigned)col       < (unsigned)IW) a0 = src[row * IW + col];
        if ((unsigned)(col + 1) < (unsigned)IW) a1 = src[row * IW + col + 1];
        v2f A  = {a0, a1};
        v2f Bv = {bandv(k0 - j), bandv(k0 + 1 - j)};  // S[k, j] = g[k - j]
        acc = __builtin_amdgcn_wmma_f32_16x16x4_f32(false, A, false, Bv,
                                                    (short)0, acc, false, false);
    }
    // C/D layout: lanes 0-15 -> M = r, lanes 16-31 -> M = 8 + r, N = lane&15
    float* dst = hb + (size_t)b * (IH * IW);
    const int orow = y0 + ((lane >> 4) << 3);
    const int ocol = x0 + (lane & 15);
#pragma unroll
    for (int r = 0; r < 8; ++r) dst[(orow + r) * IW + ocol] = acc[r];
}

// ---------------------------------------------------------------------------
// Kernel 2: vertical 5-tap blur via WMMA.
// Out(16x16) = T(16x20) * In(20x16); T[i,k] = g[k-i] is constant (A operand),
// B operand is 20 rows of hb (with zero padding at top/bottom edges).
// ---------------------------------------------------------------------------
__global__ __launch_bounds__(256) void k_vblur(const float* __restrict__ hb,
                                               float* __restrict__ vb) {
    const int lane = threadIdx.x & 31;
    const int wave = threadIdx.x >> 5;
    const int tile = blockIdx.x * 8 + wave;
    const int b    = tile >> 12;
    const int rem  = tile & 4095;
    const int y0   = (rem >> 6) << 4;
    const int x0   = (rem & 63) << 4;

    const float* src = hb + (size_t)b * (IH * IW);
    const int n  = lane & 15;             // M for A, N for B
    const int cb = (lane >> 4) << 1;

    v8f acc = {0.f, 0.f, 0.f, 0.f, 0.f, 0.f, 0.f, 0.f};
#pragma unroll
    for (int c = 0; c < 5; ++c) {
        const int ks = 4 * c + cb;
        v2f A = {bandv(ks - n), bandv(ks + 1 - n)};   // T[i,k] = g[k-i]
        const int ry = y0 - 2 + ks;
        float b0 = 0.f, b1 = 0.f;
        if ((unsigned)ry       < (unsigned)IH) b0 = src[ry * IW + x0 + n];
        if ((unsigned)(ry + 1) < (unsigned)IH) b1 = src[(ry + 1) * IW + x0 + n];
        v2f Bv = {b0, b1};
        acc = __builtin_amdgcn_wmma_f32_16x16x4_f32(false, A, false, Bv,
                                                    (short)0, acc, false, false);
    }
    float* dst = vb + (size_t)b * (IH * IW);
    const int orow = y0 + ((lane >> 4) << 3);
    const int ocol = x0 + (lane & 15);
#pragma unroll
    for (int r = 0; r < 8; ++r) dst[(orow + r) * IW + ocol] = acc[r];
}

// ---------------------------------------------------------------------------
// Kernel 3: Sobel + magnitude + orientation + directional NMS (per 16x16 tile,
// 20x20 LDS halo of the blurred image; 18x18 LDS magnitude tile), plus
// per-image max via block reduce + atomicMax (uint-bit compare, mags >= 0).
// Writes un-normalized thin edges.
// ---------------------------------------------------------------------------
__global__ __launch_bounds__(256) void k_grad(const float* __restrict__ vb,
                                              float* __restrict__ thin,
                                              unsigned* __restrict__ gmax) {
    __shared__ float s_vb[20][21];
    __shared__ float s_mag[18][19];
    __shared__ float s_red[8];

    const int tx = threadIdx.x, ty = threadIdx.y;
    const int tid = ty * 16 + tx;
    const int x0 = blockIdx.x << 4, y0 = blockIdx.y << 4;
    const int b  = blockIdx.z;
    const float* src = vb + (size_t)b * (IH * IW);

    for (int i = tid; i < 400; i += 256) {
        const int r = i / 20, c = i - r * 20;
        const int yy = y0 - 2 + r, xx = x0 - 2 + c;
        float v = 0.f;
        if ((unsigned)yy < (unsigned)IH && (unsigned)xx < (unsigned)IW)
            v = src[yy * IW + xx];
        s_vb[r][c] = v;
    }
    __syncthreads();

    for (int i = tid; i < 324; i += 256) {
        const int r = i / 18, c = i - r * 18;   // pixel (y0-1+r, x0-1+c)
        const float gxv = (s_vb[r][c]   - s_vb[r][c+2])
                  + 2.f * (s_vb[r+1][c] - s_vb[r+1][c+2])
                  +       (s_vb[r+2][c] - s_vb[r+2][c+2]);
        const float gyv = (s_vb[r][c] + 2.f * s_vb[r][c+1] + s_vb[r][c+2])
                        - (s_vb[r+2][c] + 2.f * s_vb[r+2][c+1] + s_vb[r+2][c+2]);
        float m = sqrtf(gxv * gxv + gyv * gyv);
        const int yy = y0 - 1 + r, xx = x0 - 1 + c;
        if (!((unsigned)yy < (unsigned)IH && (unsigned)xx < (unsigned)IW))
            m = 0.f;                             // dir-conv zero padding
        s_mag[r][c] = m;
    }
    __syncthreads();

    // Center pixel (y0+ty, x0+tx): s_vb index +2, s_mag index +1
    const float gxv = (s_vb[ty+1][tx+1] - s_vb[ty+1][tx+3])
              + 2.f * (s_vb[ty+2][tx+1] - s_vb[ty+2][tx+3])
              +       (s_vb[ty+3][tx+1] - s_vb[ty+3][tx+3]);
    const float gyv = (s_vb[ty+1][tx+1] + 2.f * s_vb[ty+1][tx+2] + s_vb[ty+1][tx+3])
                    - (s_vb[ty+3][tx+1] + 2.f * s_vb[ty+3][tx+2] + s_vb[ty+3][tx+3]);
    const float magc = s_mag[ty+1][tx+1];

    const float ang = atan2f(gyv, gxv) * 57.2958279f + 180.f;  // 180/3.14159
    const int ip = ((int)rintf(ang * (1.f / 45.f))) & 7;       // (round(ori/45)) % 8
    const int in_ = (ip + 4) & 7;
    // dy = {0,1,1,1,0,-1,-1,-1}, dx = {1,1,0,-1,-1,-1,0,1}, nibble-packed (+1)
    const int dyp = (int)((0x00012221u >> (4 * ip )) & 15u) - 1;
    const int dxp = (int)((0x21000122u >> (4 * ip )) & 15u) - 1;
    const int dyn = (int)((0x00012221u >> (4 * in_)) & 15u) - 1;
    const int dxn = (int)((0x21000122u >> (4 * in_)) & 15u) - 1;
    const float sp = magc - s_mag[ty + 1 + dyp][tx + 1 + dxp];
    const float sn = magc - s_mag[ty + 1 + dyn][tx + 1 + dxn];
    const float t  = (fminf(sp, sn) > 0.f) ? magc : 0.f;
    thin[(size_t)b * (IH * IW) + (y0 + ty) * IW + (x0 + tx)] = t;

    // Per-image max of grad magnitude
    float m = magc;
#pragma unroll
    for (int off = 16; off; off >>= 1) m = fmaxf(m, __shfl_down(m, off, 32));
    if ((tid & 31) == 0) s_red[tid >> 5] = m;
    __syncthreads();
    if (tid == 0) {
        float mm = s_red[0];
#pragma unroll
        for (int i = 1; i < 8; ++i) mm = fmaxf(mm, s_red[i]);
        atomicMax(&gmax[b], __float_as_uint(mm));
    }
}

// ---------------------------------------------------------------------------
// Kernel 4: normalize thin edges by 255 / per-image max.
// ---------------------------------------------------------------------------
__global__ __launch_bounds__(256) void k_norm(const float* __restrict__ thin,
                                              const unsigned* __restrict__ gmax,
                                              float* __restrict__ out, int n) {
    const int i = blockIdx.x * 256 + threadIdx.x;
    if (i >= n) return;
    const int b = i >> 20;                       // 1024*1024 = 2^20
    const float mx = __uint_as_float(gmax[b]);
    out[i] = thin[i] * (255.f / mx);
}

__global__ void k_init(unsigned* gmax) {
    if (threadIdx.x < 8) gmax[threadIdx.x] = 0u;
}

extern "C" void kernel_launch(void* const* d_in, const int* in_sizes, int n_in,
                              void* d_out, int out_size, void* d_ws, size_t ws_size,
                              hipStream_t stream) {
    const float* img = (const float*)d_in[0];
    const int nimg = in_sizes[0] / (IH * IW);    // 8
    const size_t plane = (size_t)IH * IW;

    float*    bufA = (float*)d_ws;               // hb, then thin (reused)
    float*    bufB = bufA + (size_t)nimg * plane; // vb
    unsigned* gmax = (unsigned*)(bufB + (size_t)nimg * plane);

    const int tiles  = nimg * 4096;              // 16x16 tiles
    const int blocks = tiles / 8;                // 8 waves (tiles) per 256-thread block

    k_init <<<1, 32, 0, stream>>>(gmax);
    k_hblur<<<blocks, 256, 0, stream>>>(img,  bufA);
    k_vblur<<<blocks, 256, 0, stream>>>(bufA, bufB);
    k_grad <<<dim3(IW / 16, IH / 16, nimg), dim3(16, 16), 0, stream>>>(bufB, bufA, gmax);
    const int n = nimg * (int)plane;
    k_norm <<<(n + 255) / 256, 256, 0, stream>>>(bufA, gmax, (float*)d_out, n);
}